// Exch_16527034155040
// MI455X (gfx1250) — compile-verified
//
#include <hip/hip_runtime.h>
#include <stdint.h>

// ---------------------------------------------------------------------------
// Exchange-field 7-point stencil, 256^3, f32.  Memory-bound (~402 MB min HBM
// traffic -> ~17us at 23.3 TB/s).  Strategy: async global->LDS staging of
// halo'd z-planes (ASYNCcnt-tracked GLOBAL_LOAD_ASYNC_TO_LDS_B32), 4-slot LDS
// ring, software-pipelined so the z+2 plane load overlaps the z-plane compute.
// A/E are touched once -> non-temporal so they don't evict reusable planes.
// ---------------------------------------------------------------------------

#define N 256
#define TX 32
#define TY 8
#define TXH (TX + 2) /* 34 */
#define TYH (TY + 2) /* 10 */
#define PLANE_ELEMS (TXH * TYH) /* 340 */
#define NTHREADS (TX * TY)
#define ZCHUNKS 8
#define ZSPAN (N / ZCHUNKS)

// DX = 4e-9 ;  E = (-A/DX^2) * sum_c f_c * s_c ;  -1/DX^2 = -6.25e16
#define NEG_INV_DX2 (-6.25e16f)

__device__ __forceinline__ uint32_t lds_addr_of(const void* p) {
  // gfx1250 generic LDS addresses are {SHARED_BASE[63:32], lds_off[31:0]}
  // (ISA 10.2 aperture rules): low 32 bits are exactly the wave-relative LDS
  // byte address the async-load VDST operand expects.
  return (uint32_t)(uintptr_t)p;
}

__device__ __forceinline__ void async_load_dword(uint32_t lds_byte_addr,
                                                 uint32_t glb_byte_off,
                                                 const void* glb_base) {
  // GVS form: vdst = LDS byte address, vaddr = per-lane 32-bit offset,
  // saddr = 64-bit base in an SGPR pair.  Tracked by ASYNCcnt.
  asm volatile("global_load_async_to_lds_b32 %0, %1, %2"
               :
               : "v"(lds_byte_addr), "v"(glb_byte_off), "s"(glb_base)
               : "memory");
}

__device__ __forceinline__ void wait_asynccnt0() {
  asm volatile("s_wait_asynccnt 0x0" ::: "memory");
}

// Stage one halo'd (TXH x TYH) plane of {sx, sy, sz, geo} at global z = zg
// into LDS slot `slot`.  Spin reads are clamped (OOB spin is always multiplied
// by an exactly-zero OOB geo, so any finite value is fine); geo OOB cells are
// written as 0.0f directly.
__device__ __forceinline__ void stage_plane(float (*slot)[TYH][TXH],
                                            const float* __restrict__ spin,
                                            const float* __restrict__ geo,
                                            int x0, int y0, int zg, int tid) {
  const int zc = zg < 0 ? 0 : (zg > N - 1 ? N - 1 : zg);
  const bool zin = (zg >= 0) && (zg < N);

  // spin components: clamped async loads
  for (int c = 0; c < 3; ++c) {
    const float* base = spin + (size_t)c * N * N * N;
    for (int e = tid; e < PLANE_ELEMS; e += NTHREADS) {
      const int ly = e / TXH;
      const int lx = e - ly * TXH;
      int gx = x0 - 1 + lx;
      int gy = y0 - 1 + ly;
      gx = gx < 0 ? 0 : (gx > N - 1 ? N - 1 : gx);
      gy = gy < 0 ? 0 : (gy > N - 1 ? N - 1 : gy);
      const uint32_t goff = (uint32_t)((((zc * N) + gy) * N + gx) * 4);
      async_load_dword(lds_addr_of(&slot[c][ly][lx]), goff, base);
    }
  }

  // geo: must be exactly zero outside the domain
  for (int e = tid; e < PLANE_ELEMS; e += NTHREADS) {
    const int ly = e / TXH;
    const int lx = e - ly * TXH;
    const int gx = x0 - 1 + lx;
    const int gy = y0 - 1 + ly;
    const bool inb = zin && gx >= 0 && gx < N && gy >= 0 && gy < N;
    if (inb) {
      const uint32_t goff = (uint32_t)((((zg * N) + gy) * N + gx) * 4);
      async_load_dword(lds_addr_of(&slot[3][ly][lx]), goff, geo);
    } else {
      slot[3][ly][lx] = 0.0f;
    }
  }
}

__global__ void __launch_bounds__(NTHREADS) exch_kernel(
    const float* __restrict__ spin, const float* __restrict__ geo,
    const float* __restrict__ A, float* __restrict__ E) {
  // 4-slot ring: [slot][arr][ly][lx], arr = {sx, sy, sz, geo}; 21.76 KB.
  __shared__ float smem[4][4][TYH][TXH];

  const int tid = threadIdx.x;
  const int tx = tid & (TX - 1);
  const int ty = tid >> 5;
  const int x0 = blockIdx.x * TX;
  const int y0 = blockIdx.y * TY;
  const int z0 = blockIdx.z * ZSPAN;
  const int x = x0 + tx;
  const int y = y0 + ty;

  // Prologue: stage planes z0-1, z0, z0+1.
  for (int k = -1; k <= 1; ++k) {
    const int zg = z0 + k;
    stage_plane(smem[(zg + 4) & 3], spin, geo, x0, y0, zg, tid);
  }
  wait_asynccnt0();
  __syncthreads();

  for (int z = z0; z < z0 + ZSPAN; ++z) {
    // Async-prefetch plane z+2 into the slot of plane z-2 (safe: the trailing
    // wait+barrier of the previous iteration retired all readers of it).
    stage_plane(smem[(z + 2) & 3], spin, geo, x0, y0, z + 2, tid);

    float(*PM)[TYH][TXH] = smem[(z + 3) & 3]; // plane z-1
    float(*PC)[TYH][TXH] = smem[z & 3];       // plane z
    float(*PP)[TYH][TXH] = smem[(z + 1) & 3]; // plane z+1

    const int lx = tx + 1;
    const int ly = ty + 1;
    const float gxm = PC[3][ly][lx - 1];
    const float gxp = PC[3][ly][lx + 1];
    const float gym = PC[3][ly - 1][lx];
    const float gyp = PC[3][ly + 1][lx];
    const float gzm = PM[3][ly][lx];
    const float gzp = PP[3][ly][lx];

    float dot = 0.0f;
#pragma unroll
    for (int c = 0; c < 3; ++c) {
      const float s = PC[c][ly][lx];
      const float f = (PC[c][ly][lx - 1] - s) * gxm +
                      (PC[c][ly][lx + 1] - s) * gxp +
                      (PC[c][ly - 1][lx] - s) * gym +
                      (PC[c][ly + 1][lx] - s) * gyp +
                      (PM[c][ly][lx] - s) * gzm +
                      (PP[c][ly][lx] - s) * gzp;
      dot += f * s;
    }

    const size_t idx = ((size_t)z * N + y) * N + x;
    // A and E are streamed exactly once: non-temporal so they don't evict the
    // spin/geo planes that have genuine L2 reuse (z+-1 and cross-block halos).
    const float a = __builtin_nontemporal_load(&A[idx]);
    __builtin_nontemporal_store(NEG_INV_DX2 * a * dot, &E[idx]);

    // Retire this wave's async loads, then rendezvous so every wave's z+2
    // data is in LDS (and every reader of slot (z+2)&3-as-(z-2) is done).
    wait_asynccnt0();
    __syncthreads();
  }
}

extern "C" void kernel_launch(void* const* d_in, const int* in_sizes, int n_in,
                              void* d_out, int out_size, void* d_ws,
                              size_t ws_size, hipStream_t stream) {
  (void)in_sizes;
  (void)n_in;
  (void)out_size;
  (void)d_ws;
  (void)ws_size;
  const float* spin = (const float*)d_in[0];
  const float* geo = (const float*)d_in[1];
  // d_in[2] = Ms : unused by the reference computation
  const float* A = (const float*)d_in[3];
  float* E = (float*)d_out;

  dim3 grid(N / TX, N / TY, ZCHUNKS);
  dim3 block(NTHREADS);
  hipLaunchKernelGGL(exch_kernel, grid, block, 0, stream, spin, geo, A, E);
}